// EMDLoss_4097398800461
// MI455X (gfx1250) — compile-verified
//
#include <hip/hip_runtime.h>

// ---------------------------------------------------------------------------
// EMD soft-assignment loss, fused. gfx1250 / CDNA5, wave32.
//
// d2(m,n) = |p_m|^2 + |g_n|^2 - 2<p_m,g_n> computed entirely inside one
// V_WMMA_F32_16X16X4_F32 per 16x16 tile:
//   A(16x4) row m = (-2px,-2py,-2pz, 1)         (constant per strip)
//   B(4x16) col n = ( gx,  gy,  gz, |g|^2)      (one b64 load per lane/tile)
//   C(16x16)      = |p_m|^2 broadcast            (constant per strip)
// Per element: d = sqrt(|d2|)  (fabs = free src modifier; |d2| only differs
// from max(d2,0) when d2 is a rounding-level negative, where both are ~0);
// e = exp2_raw(d * -1/(T*ln2)); s += e; w += e*d.
// Fixed-shift softmin => plain sums => associative => deterministic tree
// reductions, no float atomics.
// ---------------------------------------------------------------------------

typedef float v2f __attribute__((ext_vector_type(2)));
typedef float v8f __attribute__((ext_vector_type(8)));

#define B_   8
#define M_   4096
#define N_   4096
#define BM_  (B_ * M_)          // 32768 rows
#define BN_  (B_ * N_)          // 32768 gt points
#define NCHUNK 4
#define TILES_PER_CHUNK ((N_ / NCHUNK) / 16)   // 64
#define KEXP (-14.426950408889634f)            // -1/(0.1 * ln2)

// workspace layout (float offsets)
#define OFF_G4 ((size_t)0)                         // BN*4 : (gx,gy,gz,|g|^2)
#define OFF_P2 (OFF_G4 + (size_t)BN_ * 4)          // BM   : |p|^2
#define OFF_S  (OFF_P2 + (size_t)BM_)              // NCHUNK*BM partial sums s
#define OFF_W  (OFF_S  + (size_t)NCHUNK * BM_)     // NCHUNK*BM partial sums w
#define OFF_PB (OFF_W  + (size_t)NCHUNK * BM_)     // 128 block partials

// --------------------------- kernel 0: prepass -----------------------------
__global__ __launch_bounds__(256) void emd_prep(const float* __restrict__ pred,
                                                const float* __restrict__ gt,
                                                float* __restrict__ ws) {
  int i = blockIdx.x * 256 + threadIdx.x;          // 0 .. 32767 (BN_==BM_)
  float* g4 = ws + OFF_G4;
  float* p2 = ws + OFF_P2;
  float gx = gt[3 * i + 0], gy = gt[3 * i + 1], gz = gt[3 * i + 2];
  g4[4 * (size_t)i + 0] = gx;
  g4[4 * (size_t)i + 1] = gy;
  g4[4 * (size_t)i + 2] = gz;
  g4[4 * (size_t)i + 3] = gx * gx + gy * gy + gz * gz;
  float px = pred[3 * i + 0], py = pred[3 * i + 1], pz = pred[3 * i + 2];
  p2[i] = px * px + py * py + pz * pz;
}

// ------------------------ kernel 1: fused WMMA core ------------------------
__global__ __launch_bounds__(256) void emd_main(const float* __restrict__ pred,
                                                float* __restrict__ ws) {
  const float*  p2    = ws + OFF_P2;
  const float2* g4h   = (const float2*)(ws + OFF_G4);  // 2 halves per point
  float*        sPart = ws + OFF_S;
  float*        wPart = ws + OFF_W;

  int tid   = threadIdx.x;
  int lane  = tid & 31;
  int l     = lane & 15;          // position within half-wave
  int half  = lane >> 4;          // 0: K={0,1}/rows 0-7 ; 1: K={2,3}/rows 8-15
  int waveId = blockIdx.x * 8 + (tid >> 5);
  int chunk  = waveId & (NCHUNK - 1);
  int strip  = waveId >> 2;                       // 0 .. 2047
  int b      = strip >> 8;                        // 256 strips per batch
  int row0   = strip << 4;                        // global row base in [0,BM)
  int nBase  = b * N_ + chunk * (N_ / NCHUNK);

  // A fragment: row = row0 + l for both halves (32-bit A 16x4 layout)
  const float* pp = pred + (size_t)(row0 + l) * 3;
  v2f a;
  if (half == 0) { a.x = -2.0f * pp[0]; a.y = -2.0f * pp[1]; }
  else           { a.x = -2.0f * pp[2]; a.y = 1.0f;          }

  // C fragment: c[i] = |p_(row0 + 8*half + i)|^2, broadcast across N lanes
  v8f cf;
#pragma unroll
  for (int i = 0; i < 8; ++i) cf[i] = p2[row0 + 8 * half + i];

  float s[8], w[8];
#pragma unroll
  for (int i = 0; i < 8; ++i) { s[i] = 0.0f; w[i] = 0.0f; }

  for (int t = 0; t < TILES_PER_CHUNK; ++t) {
    int n = nBase + t * 16 + l;
    float2 bv = g4h[(size_t)n * 2 + half];        // (gx,gy) or (gz,|g|^2)
    v2f bb; bb.x = bv.x; bb.y = bv.y;

    // d2 tile = A*B + C  (8 args: neg_a, A, neg_b, B, c_mod, C, reuse_a/b)
    v8f d2 = __builtin_amdgcn_wmma_f32_16x16x4_f32(
        false, a, false, bb, (short)0, cf, false, false);

#pragma unroll
    for (int i = 0; i < 8; ++i) {
      // fabs folds into v_sqrt_f32 |src| modifier -> zero-cost clamp
      float d = __builtin_amdgcn_sqrtf(__builtin_fabsf(d2[i]));
      float e = __builtin_amdgcn_exp2f(d * KEXP); // exp(-d/T), fixed shift 0
      s[i] += e;
      w[i] += e * d;
    }
  }

  // reduce across the 16 lanes that share each row (xor net, fixed order)
#pragma unroll
  for (int i = 0; i < 8; ++i) {
#pragma unroll
    for (int off = 1; off < 16; off <<= 1) {
      s[i] += __shfl_xor(s[i], off, 32);
      w[i] += __shfl_xor(w[i], off, 32);
    }
  }

  if (l == 0) {                                   // lanes 0 and 16 write
    size_t base = (size_t)chunk * BM_ + row0 + 8 * half;
#pragma unroll
    for (int i = 0; i < 8; ++i) {
      sPart[base + i] = s[i];
      wPart[base + i] = w[i];
    }
  }
}

// ---------------- kernel 2: per-row w/s, block tree reduction --------------
__global__ __launch_bounds__(256) void emd_rows(float* __restrict__ ws) {
  const float* sPart = ws + OFF_S;
  const float* wPart = ws + OFF_W;
  float*       pb    = ws + OFF_PB;
  __shared__ float sm[256];

  int tid = threadIdx.x;
  int row = blockIdx.x * 256 + tid;               // 128 blocks cover BM_
  float s = 0.0f, w = 0.0f;
#pragma unroll
  for (int c = 0; c < NCHUNK; ++c) {              // fixed summation order
    s += sPart[(size_t)c * BM_ + row];
    w += wPart[(size_t)c * BM_ + row];
  }
  sm[tid] = (s > 0.0f) ? (w / s) : 0.0f;
  __syncthreads();
#pragma unroll
  for (int st = 128; st > 0; st >>= 1) {
    if (tid < st) sm[tid] += sm[tid + st];
    __syncthreads();
  }
  if (tid == 0) pb[blockIdx.x] = sm[0];
}

// ----------------------- kernel 3: final scalar ----------------------------
__global__ __launch_bounds__(128) void emd_final(const float* __restrict__ ws,
                                                 float* __restrict__ out) {
  __shared__ float sm[128];
  int tid = threadIdx.x;
  sm[tid] = ws[OFF_PB + tid];
  __syncthreads();
#pragma unroll
  for (int st = 64; st > 0; st >>= 1) {
    if (tid < st) sm[tid] += sm[tid + st];
    __syncthreads();
  }
  if (tid == 0) out[0] = sm[0] * (1.0f / (float)BM_);   // mean over B*M
}

// ---------------------------------------------------------------------------
extern "C" void kernel_launch(void* const* d_in, const int* in_sizes, int n_in,
                              void* d_out, int out_size, void* d_ws, size_t ws_size,
                              hipStream_t stream) {
  (void)in_sizes; (void)n_in; (void)out_size; (void)ws_size;
  const float* pred = (const float*)d_in[0];   // [8,4096,3] f32
  const float* gt   = (const float*)d_in[1];   // [8,4096,3] f32
  float* out = (float*)d_out;
  float* ws  = (float*)d_ws;

  emd_prep <<<BN_ / 256, 256, 0, stream>>>(pred, gt, ws);
  emd_main <<<(2048 * NCHUNK) / 8, 256, 0, stream>>>(pred, ws); // 8192 waves
  emd_rows <<<BM_ / 256, 256, 0, stream>>>(ws);
  emd_final<<<1, 128, 0, stream>>>(ws, out);
}